// ShiftMatch_77240691851439
// MI455X (gfx1250) — compile-verified
//
#include <hip/hip_runtime.h>

// ============================================================================
// ShiftMatch on MI455X (gfx1250): per-channel whiten-recolor along H then W.
// eigh is replaced by Newton-Schulz iteration (sqrt & inv-sqrt via GEMMs),
// all GEMMs run on V_WMMA_F32_16X16X4_F32 (fp32 WMMA, wave32).
// ============================================================================

typedef __attribute__((ext_vector_type(2))) float v2f;
typedef __attribute__((ext_vector_type(8))) float v8f;

#define CCH     256
#define DIMV    56          // valid H / W length
#define DP      64          // padded dimension
#define MROWS   3584        // N*W == N*H == 64*56
#define LSTRIDE 65          // LDS row stride (bank-conflict pad)
#define NSITER  20          // Newton-Schulz iterations

__device__ __forceinline__ v8f wmma4(v2f a, v2f b, v8f c) {
  // D(16x16,f32) = A(16x4,f32) x B(4x16,f32) + C
  return __builtin_amdgcn_wmma_f32_16x16x4_f32(
      false, a, false, b, (short)0, c, false, false);
}

// D = alpha * (A @ B), 64x64 matrices in LDS with row stride LSTRIDE.
// 8 waves; wave wv owns tiles 2wv and 2wv+1 of the 4x4 16x16-tile grid.
// Caller must __syncthreads() before (inputs ready) and after (before reuse).
__device__ __forceinline__ void gemm64(const float* A, const float* B,
                                       float* D, float alpha) {
  const int lane = threadIdx.x & 31;
  const int wv   = threadIdx.x >> 5;
  const int lr   = lane & 15;
  const int lh   = lane >> 4;
  for (int t = 2 * wv; t <= 2 * wv + 1; ++t) {
    const int i0 = (t >> 2) << 4;
    const int j0 = (t & 3) << 4;
    v8f acc;
#pragma unroll
    for (int r = 0; r < 8; ++r) acc[r] = 0.0f;
#pragma unroll
    for (int kk = 0; kk < 16; ++kk) {
      const int kb = kk * 4 + 2 * lh;
      v2f a, b;
      a.x = A[(i0 + lr) * LSTRIDE + kb];       // A[row][k]
      a.y = A[(i0 + lr) * LSTRIDE + kb + 1];
      b.x = B[kb * LSTRIDE + j0 + lr];         // B[k][col]
      b.y = B[(kb + 1) * LSTRIDE + j0 + lr];
      acc = wmma4(a, b, acc);
    }
#pragma unroll
    for (int r = 0; r < 8; ++r)
      D[(i0 + r + 8 * lh) * LSTRIDE + j0 + lr] = alpha * acc[r];
  }
}

__device__ __forceinline__ float reduce256(float v, float* red) {
  red[threadIdx.x] = v;
  __syncthreads();
  for (int s = 128; s > 0; s >>= 1) {
    if (threadIdx.x < (unsigned)s) red[threadIdx.x] += red[threadIdx.x + s];
    __syncthreads();
  }
  float r = red[0];
  __syncthreads();
  return r;
}

// Newton-Schulz: on entry Y holds SPD matrix A (identity-padded, 64x64 LDS).
// On exit (pointers rotated): Y = sqrt(A/f), Z = (A/f)^(-1/2), fro = f.
// sqrt(A) = Y*sqrt(f); A^(-1/2) = Z/sqrt(f).
__device__ void newton_schulz(float*& Y, float*& Z, float*& T, float*& S,
                              float* red, float& froOut) {
  const int j  = threadIdx.x & 63;
  const int ib = threadIdx.x >> 6;
  float p = 0.0f;
#pragma unroll
  for (int k = 0; k < 16; ++k) {
    float v = Y[(ib + 4 * k) * LSTRIDE + j];
    p += v * v;
  }
  float fro  = sqrtf(reduce256(p, red));
  float rinv = 1.0f / fro;
#pragma unroll
  for (int k = 0; k < 16; ++k) {
    int i = ib + 4 * k;
    Y[i * LSTRIDE + j] *= rinv;
    Z[i * LSTRIDE + j] = (i == j) ? 1.0f : 0.0f;
  }
  __syncthreads();
  for (int it = 0; it < NSITER; ++it) {
    gemm64(Z, Y, T, 1.0f);
    __syncthreads();
#pragma unroll
    for (int k = 0; k < 16; ++k) {
      int i = ib + 4 * k;
      float v = T[i * LSTRIDE + j];
      T[i * LSTRIDE + j] = ((i == j) ? 1.5f : 0.0f) - 0.5f * v;
    }
    __syncthreads();
    gemm64(Y, T, S, 1.0f);   // Ynew in S
    __syncthreads();
    gemm64(T, Z, Y, 1.0f);   // Znew in old-Y buffer (old Y dead)
    __syncthreads();
    float* oy = Y; float* oz = Z;
    Y = S; Z = oy; S = oz;   // rotate buffers
  }
  froOut = fro;
}

// ---------------------------------------------------------------------------
// K0: pack x(N,H,W,C) -> XH[c][m=n*56+w][hp] (hp=hh, zero-padded to 64)
// ---------------------------------------------------------------------------
__global__ __launch_bounds__(256) void k_pack(const float* __restrict__ x,
                                              float* __restrict__ XH) {
  __shared__ float tile[DP * LSTRIDE];
  const int s  = blockIdx.x;        // n*56 + w
  const int n  = s / DIMV;
  const int w  = s % DIMV;
  const int c0 = blockIdx.y * 64;
  const int j  = threadIdx.x & 63;
  const int ib = threadIdx.x >> 6;
#pragma unroll
  for (int k = 0; k < 16; ++k) {
    int i = ib + 4 * k;             // hh
    if (i < DIMV)
      tile[i * LSTRIDE + j] =
          x[(((size_t)n * DIMV + i) * DIMV + w) * CCH + c0 + j];
  }
  __syncthreads();
#pragma unroll
  for (int k = 0; k < 16; ++k) {
    int jc = ib + 4 * k;            // channel-local
    int i  = j;                     // hh (contiguous in XH)
    float v = (i < DIMV) ? tile[i * LSTRIDE + jc] : 0.0f;
    XH[(((size_t)(c0 + jc)) * MROWS + (size_t)n * DIMV + w) * DP + i] = v;
  }
}

// ---------------------------------------------------------------------------
// K1: per-channel stats + Newton-Schulz + T,b   (one workgroup per channel)
// ---------------------------------------------------------------------------
__global__ __launch_bounds__(256) void k_stats(
    const float* __restrict__ Xbuf, const float* __restrict__ covIn,
    const float* __restrict__ muIn, const int* __restrict__ cntPtr,
    float* __restrict__ Tout, float* __restrict__ bOut) {
  __shared__ float B0[DP * LSTRIDE], B1[DP * LSTRIDE], B2[DP * LSTRIDE],
                   B3[DP * LSTRIDE], B4[DP * LSTRIDE], B5[DP * LSTRIDE];
  __shared__ float red[256], muT[DP], muTr[DP];
  const int c    = blockIdx.x;
  const int tid  = threadIdx.x;
  const int j    = tid & 63;
  const int ib   = tid >> 6;
  const int lane = tid & 31;
  const int wv   = tid >> 5;
  const int lr   = lane & 15;
  const int lh   = lane >> 4;
  const float* Xc = Xbuf + (size_t)c * MROWS * DP;
  const float invM = 1.0f / (float)MROWS;

  // ---- Gram (raw second moment, padded cols are zero) + column sums ----
  v8f acc0, acc1;
#pragma unroll
  for (int r = 0; r < 8; ++r) { acc0[r] = 0.0f; acc1[r] = 0.0f; }
  float ssum = 0.0f;
  for (int ch = 0; ch < MROWS / DP; ++ch) {
    const float* src = Xc + (size_t)ch * DP * DP;
#pragma unroll
    for (int k = 0; k < 16; ++k) {
      int row = ib + 4 * k;
      float v = src[(size_t)row * DP + j];
      B0[row * LSTRIDE + j] = v;
      ssum += v;
    }
    __syncthreads();
    for (int tsel = 0; tsel < 2; ++tsel) {
      const int t  = 2 * wv + tsel;
      const int i0 = (t >> 2) << 4;
      const int j0 = (t & 3) << 4;
      v8f acc = tsel ? acc1 : acc0;
#pragma unroll
      for (int kk = 0; kk < 16; ++kk) {
        const int kb = kk * 4 + 2 * lh;
        v2f a, b;
        a.x = B0[kb * LSTRIDE + i0 + lr];        // A = chunk^T
        a.y = B0[(kb + 1) * LSTRIDE + i0 + lr];
        b.x = B0[kb * LSTRIDE + j0 + lr];
        b.y = B0[(kb + 1) * LSTRIDE + j0 + lr];
        acc = wmma4(a, b, acc);
      }
      if (tsel) acc1 = acc; else acc0 = acc;
    }
    __syncthreads();
  }
  // mean vector
  red[tid] = ssum;
  __syncthreads();
  if (tid < 64) {
    float s = red[tid] + red[tid + 64] + red[tid + 128] + red[tid + 192];
    muT[tid] = s * invM;
  }
  // dump Gram into B1
  for (int tsel = 0; tsel < 2; ++tsel) {
    const int t  = 2 * wv + tsel;
    const int i0 = (t >> 2) << 4;
    const int j0 = (t & 3) << 4;
    v8f acc = tsel ? acc1 : acc0;
#pragma unroll
    for (int r = 0; r < 8; ++r)
      B1[(i0 + r + 8 * lh) * LSTRIDE + j0 + lr] = acc[r];
  }
  if (tid < 64) muTr[tid] = (tid < DIMV) ? muIn[c * DIMV + tid] : 0.0f;
  __syncthreads();

  // ---- covTest (B1) with diag jitter; covTrain (B2); identity padding ----
  const float invCnt = 1.0f / (float)(*cntPtr);
#pragma unroll
  for (int k = 0; k < 16; ++k) {
    int i = ib + 4 * k;
    float ct, cv;
    if (i < DIMV && j < DIMV) {
      ct = B1[i * LSTRIDE + j] * invM - muT[i] * muT[j];
      if (i == j) ct *= 1.001f;
      cv = covIn[((size_t)c * DIMV + i) * DIMV + j] * invCnt -
           muTr[i] * muTr[j];
    } else {
      ct = (i == j) ? 1.0f : 0.0f;
      cv = (i == j) ? 1.0f : 0.0f;
    }
    B1[i * LSTRIDE + j] = ct;
    B2[i * LSTRIDE + j] = cv;
  }
  __syncthreads();

  // ---- NS on covTrain -> Y1 ~ sqrt(cov_train / f1) ----
  float *Y = B2, *Z = B3, *T = B4, *S = B5;
  float fro1;
  newton_schulz(Y, Z, T, S, red, fro1);
#pragma unroll
  for (int k = 0; k < 16; ++k) {
    int i = ib + 4 * k;
    B0[i * LSTRIDE + j] = Y[i * LSTRIDE + j];   // stash scaled sqrt_train
  }
  __syncthreads();

  // ---- NS on covTest -> Z2 ~ (cov_test / f2)^(-1/2) ----
  float *Y2 = B1, *Z2 = B2, *T2 = B3, *S2 = B4;
  float fro2;
  newton_schulz(Y2, Z2, T2, S2, red, fro2);

  // ---- T = invsqrt(cov_test) @ sqrt(cov_train) = (Z2 @ Y1) * sqrt(f1/f2) --
  const float alpha = sqrtf(fro1) / sqrtf(fro2);
  gemm64(Z2, B0, B5, alpha);
  __syncthreads();

  // bias b = mu_train - mu_test @ T ; store T, b
  if (tid < 64) {
    float s = 0.0f;
    for (int d = 0; d < 64; ++d) s += muT[d] * B5[d * LSTRIDE + tid];
    bOut[c * DP + tid] = muTr[tid] - s;
  }
#pragma unroll
  for (int k = 0; k < 16; ++k) {
    int i = ib + 4 * k;
    Tout[((size_t)c * DP + i) * DP + j] = B5[i * LSTRIDE + j];
  }
}

// ---------------------------------------------------------------------------
// K2: apply O = chunk @ T + b.  One block per (c, n), 56 valid rows (+8 pad).
// transposeOut=1 (H phase): out[c][n*56+hh][w] = O[w][hh]  (axis swap H->W)
// transposeOut=0 (W phase): out[c][n*56+hh][w] = O[hh][w]
// ---------------------------------------------------------------------------
__global__ __launch_bounds__(256) void k_apply(
    const float* __restrict__ Xbuf, const float* __restrict__ Tin,
    const float* __restrict__ bIn, float* __restrict__ Obuf,
    int transposeOut) {
  __shared__ float Tl[DP * LSTRIDE], Cl[DP * LSTRIDE];
  __shared__ float bl[DP];
  const int c    = blockIdx.x;
  const int n    = blockIdx.y;
  const int tid  = threadIdx.x;
  const int j    = tid & 63;
  const int ib   = tid >> 6;
  const int lane = tid & 31;
  const int wv   = tid >> 5;
  const int lr   = lane & 15;
  const int lh   = lane >> 4;

#pragma unroll
  for (int k = 0; k < 16; ++k) {
    int i = ib + 4 * k;
    Tl[i * LSTRIDE + j] = Tin[((size_t)c * DP + i) * DP + j];
  }
  if (tid < 64) bl[tid] = bIn[c * DP + tid];
  const float* Xc = Xbuf + (size_t)c * MROWS * DP;
#pragma unroll
  for (int k = 0; k < 16; ++k) {
    int r = ib + 4 * k;
    int m = n * DIMV + r;
    if (m > MROWS - 1) m = MROWS - 1;   // clamp pad rows (never written out)
    Cl[r * LSTRIDE + j] = Xc[(size_t)m * DP + j];
  }
  __syncthreads();

  v8f acc0, acc1;
  for (int tsel = 0; tsel < 2; ++tsel) {
    const int t  = 2 * wv + tsel;
    const int i0 = (t >> 2) << 4;
    const int j0 = (t & 3) << 4;
    v8f acc;
#pragma unroll
    for (int r = 0; r < 8; ++r) acc[r] = 0.0f;
#pragma unroll
    for (int kk = 0; kk < 16; ++kk) {
      const int kb = kk * 4 + 2 * lh;
      v2f a, b;
      a.x = Cl[(i0 + lr) * LSTRIDE + kb];
      a.y = Cl[(i0 + lr) * LSTRIDE + kb + 1];
      b.x = Tl[kb * LSTRIDE + j0 + lr];
      b.y = Tl[(kb + 1) * LSTRIDE + j0 + lr];
      acc = wmma4(a, b, acc);
    }
    const float bb = bl[j0 + lr];
#pragma unroll
    for (int r = 0; r < 8; ++r) acc[r] += bb;
    if (tsel) acc1 = acc; else acc0 = acc;
  }
  __syncthreads();
  for (int tsel = 0; tsel < 2; ++tsel) {
    const int t  = 2 * wv + tsel;
    const int i0 = (t >> 2) << 4;
    const int j0 = (t & 3) << 4;
    v8f acc = tsel ? acc1 : acc0;
#pragma unroll
    for (int r = 0; r < 8; ++r)
      Cl[(i0 + r + 8 * lh) * LSTRIDE + j0 + lr] = acc[r];
  }
  __syncthreads();

  float* dst = Obuf + ((size_t)c * MROWS + (size_t)n * DIMV) * DP;
#pragma unroll
  for (int k = 0; k < 16; ++k) {
    int r = ib + 4 * k;          // output row within block
    if (r >= DIMV) continue;
    int col = j;
    float v;
    if (transposeOut) v = (col < DIMV) ? Cl[col * LSTRIDE + r] : 0.0f;
    else              v = (col < DIMV) ? Cl[r * LSTRIDE + col] : 0.0f;
    dst[(size_t)r * DP + col] = v;
  }
}

// ---------------------------------------------------------------------------
// K3: unpack XO[c][n*56+hh][w] -> out(N,H,W,C)
// ---------------------------------------------------------------------------
__global__ __launch_bounds__(256) void k_unpack(const float* __restrict__ XO,
                                                float* __restrict__ out) {
  __shared__ float tile[DP * LSTRIDE];
  const int s  = blockIdx.x;       // n*56 + hh
  const int c0 = blockIdx.y * 64;
  const int tid = threadIdx.x;
  const int i  = tid & 63;         // read: w (contiguous in XO)
  const int ib = tid >> 6;
#pragma unroll
  for (int k = 0; k < 16; ++k) {
    int jc = ib + 4 * k;           // channel-local
    tile[jc * LSTRIDE + i] =
        XO[(((size_t)(c0 + jc)) * MROWS + s) * DP + i];
  }
  __syncthreads();
#pragma unroll
  for (int k = 0; k < 16; ++k) {
    int w = ib + 4 * k;
    if (w < DIMV)
      out[((size_t)s * DIMV + w) * CCH + c0 + i] = tile[i * LSTRIDE + w];
  }
}

// ---------------------------------------------------------------------------
extern "C" void kernel_launch(void* const* d_in, const int* in_sizes, int n_in,
                              void* d_out, int out_size, void* d_ws,
                              size_t ws_size, hipStream_t stream) {
  (void)in_sizes; (void)n_in; (void)out_size; (void)ws_size;
  const float* x    = (const float*)d_in[0];
  const float* covH = (const float*)d_in[1];
  const float* muH  = (const float*)d_in[2];
  const float* covW = (const float*)d_in[3];
  const float* muW  = (const float*)d_in[4];
  const int*   cnt  = (const int*)d_in[5];
  float* out = (float*)d_out;
  float* ws  = (float*)d_ws;

  const size_t matSz = (size_t)CCH * MROWS * DP;   // 58,720,256 floats
  float* XH = ws;                                  // H-axis matrices
  float* XW = XH + matSz;                          // W-axis matrices
  float* TH = XW + matSz;                          // (C,64,64)
  float* bH = TH + (size_t)CCH * DP * DP;
  float* TW = bH + (size_t)CCH * DP;
  float* bW = TW + (size_t)CCH * DP * DP;
  // total workspace: ~478 MB

  dim3 blk(256);
  k_pack  <<<dim3(MROWS, 4),  blk, 0, stream>>>(x, XH);
  k_stats <<<dim3(CCH),       blk, 0, stream>>>(XH, covH, muH, cnt, TH, bH);
  k_apply <<<dim3(CCH, 64),   blk, 0, stream>>>(XH, TH, bH, XW, 1);
  k_stats <<<dim3(CCH),       blk, 0, stream>>>(XW, covW, muW, cnt, TW, bW);
  k_apply <<<dim3(CCH, 64),   blk, 0, stream>>>(XW, TW, bW, XH, 0); // XO=XH
  k_unpack<<<dim3(MROWS, 4),  blk, 0, stream>>>(XH, out);
}